// RelativePosTransformerLayer_10445360464471
// MI455X (gfx1250) — compile-verified
//
#include <hip/hip_runtime.h>
#include <hip/hip_bf16.h>
#include <math.h>

typedef __attribute__((ext_vector_type(2))) float v2f;
typedef __attribute__((ext_vector_type(8))) float v8f;
typedef __attribute__((ext_vector_type(4))) int   v4i;

// Problem constants (reference: S=96, N=8, E=512, H=8, d=64, F=2048)
static constexpr int cS = 96;
static constexpr int cN = 8;
static constexpr int cE = 512;
static constexpr int cH = 8;
static constexpr int cD = 64;
static constexpr int cF = 2048;
static constexpr int cM = cS * cN;          // 768 flattened rows
#define NEG_MASK (-103.616329f)             // log(1e-45)

#if defined(__has_builtin)
#if __has_builtin(__builtin_amdgcn_global_load_async_to_lds_b128)
#define HAVE_ASYNC_LDS 1
#endif
#endif

// ---------------------------------------------------------------------------
// Generic fp32 WMMA GEMM:  C[M,N] = A[M,K] (row-major) * B[N,K]^T + bias[N]
// Each wave computes a 16(M) x 64(N) tile using V_WMMA_F32_16X16X4_F32.
// Software-pipelined: fragments for step k+4 are loaded into fresh registers
// before the 4 WMMAs of step k, so global loads overlap the matrix pipe.
// Requires: M%16==0, N%64==0, K%8==0. Uniform per-wave tile -> EXEC all ones.
// ---------------------------------------------------------------------------
__global__ void gemm_nt_wmma(const float* __restrict__ A, const float* __restrict__ B,
                             const float* __restrict__ bias, float* __restrict__ C,
                             int M, int N, int K, int relu)
{
    const int lane = threadIdx.x & 31;
    const int tile = blockIdx.x * (blockDim.x >> 5) + (threadIdx.x >> 5);
    const int nGroups = N >> 6;
    const int mTiles  = M >> 4;
    if (tile >= mTiles * nGroups) return;
    const int tm = tile / nGroups;
    const int tn = tile % nGroups;
    const int m0 = tm << 4;
    const int n0 = tn << 6;
    const int l16 = lane & 15;
    const int hi  = lane >> 4;            // K half (0: K=0,1  1: K=2,3)
    const int ka0 = hi << 1;

    const float* Arow = A + (size_t)(m0 + l16) * K + ka0;
    const float* B0   = B + (size_t)(n0 + l16) * K + ka0;

    v8f acc[4] = {};
    v2f a_cur = *(const v2f*)(Arow);
    v2f b_cur[4];
#pragma unroll
    for (int t = 0; t < 4; ++t)
        b_cur[t] = *(const v2f*)(B0 + (size_t)(t << 4) * K);

    for (int k0 = 0; k0 < K - 4; k0 += 4) {
        // prefetch next step's fragments into fresh registers
        v2f a_nxt = *(const v2f*)(Arow + k0 + 4);
        v2f b_nxt[4];
#pragma unroll
        for (int t = 0; t < 4; ++t)
            b_nxt[t] = *(const v2f*)(B0 + (size_t)(t << 4) * K + k0 + 4);
#pragma unroll
        for (int t = 0; t < 4; ++t)
            acc[t] = __builtin_amdgcn_wmma_f32_16x16x4_f32(
                false, a_cur, false, b_cur[t], (short)0, acc[t], false, false);
        a_cur = a_nxt;
#pragma unroll
        for (int t = 0; t < 4; ++t) b_cur[t] = b_nxt[t];
    }
#pragma unroll
    for (int t = 0; t < 4; ++t)
        acc[t] = __builtin_amdgcn_wmma_f32_16x16x4_f32(
            false, a_cur, false, b_cur[t], (short)0, acc[t], false, false);

#pragma unroll
    for (int t = 0; t < 4; ++t) {
        const int col = n0 + (t << 4) + l16;
        const float bs = bias[col];
#pragma unroll
        for (int r = 0; r < 8; ++r) {
            const int m = m0 + r + (hi << 3);
            float v = acc[t][r] + bs;
            if (relu) v = fmaxf(v, 0.0f);
            C[(size_t)m * N + col] = v;
        }
    }
}

// ---------------------------------------------------------------------------
// Attention logits, dense part: attn[n,h,i,j] = sum_d q[i,n,h,d]*k[j,n,h,d]
// q = qk[:, :, 0:512], k = qk[:, :, 512:1024]. Wave computes 16(i) x 96(j),
// K=64, via 6 accumulator tiles, software-pipelined like the GEMM.
// ---------------------------------------------------------------------------
__global__ void qk_scores_wmma(const float* __restrict__ qk, float* __restrict__ attn)
{
    const int lane = threadIdx.x & 31;
    const int wid  = blockIdx.x * (blockDim.x >> 5) + (threadIdx.x >> 5);
    if (wid >= cN * cH * (cS / 16)) return;   // 64 * 6
    const int tm = wid % 6;
    const int nh = wid / 6;
    const int h  = nh & 7;
    const int n  = nh >> 3;
    const int l16 = lane & 15;
    const int hi  = lane >> 4;
    const int ka0 = hi << 1;

    // q row i = tm*16 + l16 ; row stride (i) = N*2E = 8192 floats
    const float* qbase = qk + ((size_t)((tm * 16 + l16) * cN + n)) * (2 * cE) + h * cD + ka0;
    // k row j = jt*16 + l16 ; jt stride = 16*8192 = 131072
    const float* kbase = qk + ((size_t)(l16 * cN + n)) * (2 * cE) + cE + h * cD + ka0;

    v8f acc[6] = {};
    v2f a_cur = *(const v2f*)(qbase);
    v2f b_cur[6];
#pragma unroll
    for (int jt = 0; jt < 6; ++jt)
        b_cur[jt] = *(const v2f*)(kbase + (size_t)jt * 131072);

#pragma unroll
    for (int k0 = 0; k0 < cD - 4; k0 += 4) {
        v2f a_nxt = *(const v2f*)(qbase + k0 + 4);
        v2f b_nxt[6];
#pragma unroll
        for (int jt = 0; jt < 6; ++jt)
            b_nxt[jt] = *(const v2f*)(kbase + (size_t)jt * 131072 + k0 + 4);
#pragma unroll
        for (int jt = 0; jt < 6; ++jt)
            acc[jt] = __builtin_amdgcn_wmma_f32_16x16x4_f32(
                false, a_cur, false, b_cur[jt], (short)0, acc[jt], false, false);
        a_cur = a_nxt;
#pragma unroll
        for (int jt = 0; jt < 6; ++jt) b_cur[jt] = b_nxt[jt];
    }
#pragma unroll
    for (int jt = 0; jt < 6; ++jt)
        acc[jt] = __builtin_amdgcn_wmma_f32_16x16x4_f32(
            false, a_cur, false, b_cur[jt], (short)0, acc[jt], false, false);

#pragma unroll
    for (int jt = 0; jt < 6; ++jt) {
#pragma unroll
        for (int r = 0; r < 8; ++r) {
            const int i = tm * 16 + r + (hi << 3);
            attn[((size_t)nh * cS + i) * cS + jt * 16 + l16] = acc[jt][r];
        }
    }
}

// ---------------------------------------------------------------------------
// Fused edge-logit + scale + mask + softmax. One block per (n,h,i).
// attn[n,h,i,j] = softmax_j( (attn_qk + sum_d q[i,d]*ek[i,j,d]) * d^-1/2, mask )
// Each thread j reads a 256B (64-float) contiguous ek row: full cachelines.
// ---------------------------------------------------------------------------
__global__ void edge_softmax_kernel(const float* __restrict__ qk,
                                    const float* __restrict__ edge_k,
                                    const unsigned char* __restrict__ mask,
                                    float* __restrict__ attn)
{
    const int bid = blockIdx.x;          // ((n*H + h)*S + i)
    const int i  = bid % cS;
    const int nh = bid / cS;
    const int h  = nh & 7;
    const int n  = nh >> 3;
    const int t  = threadIdx.x;          // blockDim = 128

    __shared__ __align__(16) float qs[cD];
    __shared__ float sv[128];
    __shared__ float red[128];

    if (t < cD)
        qs[t] = qk[((size_t)(i * cN + n)) * (2 * cE) + h * cD + t];
    __syncthreads();

    const size_t arow = ((size_t)nh * cS + i) * cS;
    float s;
    if (t < cS) {
        const float4* ek4 = (const float4*)(edge_k +
            ((size_t)(i * cS + t) * cN + n) * cE + h * cD);
        const float4* q4 = (const float4*)qs;
        float acc = 0.0f;
#pragma unroll
        for (int u = 0; u < cD / 4; ++u) {
            float4 e  = ek4[u];
            float4 qv = q4[u];
            acc += e.x * qv.x + e.y * qv.y + e.z * qv.z + e.w * qv.w;
        }
        s = (attn[arow + t] + acc) * 0.125f;   // 1/sqrt(64)
        if (mask[((size_t)n * cS + i) * cS + t]) s = NEG_MASK;
    } else {
        s = -INFINITY;
    }
    sv[t] = s;
    red[t] = s;
    __syncthreads();
    for (int off = 64; off > 0; off >>= 1) {
        if (t < off) red[t] = fmaxf(red[t], red[t + off]);
        __syncthreads();
    }
    const float mx = red[0];
    __syncthreads();
    const float e = (t < cS) ? __expf(sv[t] - mx) : 0.0f;
    red[t] = e;
    __syncthreads();
    for (int off = 64; off > 0; off >>= 1) {
        if (t < off) red[t] += red[t + off];
        __syncthreads();
    }
    const float inv = 1.0f / red[0];
    if (t < cS) attn[arow + t] = e * inv;
}

// ---------------------------------------------------------------------------
// out[i,n,h,d] = sum_j attn[n,h,i,j] * (v[j,n,h,d] + ev[i,j,n,h,d])
// One block (64 threads over d) per (n,h,i). This kernel owns the 151 MB
// edge_v stream. If the async-to-LDS builtin exists, edge_v is staged through
// LDS with GLOBAL_LOAD_ASYNC_TO_LDS_B128 (double-buffered 4-row stages, one
// async instruction per wave per stage, tracked with ASYNCcnt); otherwise a
// direct coalesced loop with global_prefetch is used.
// ---------------------------------------------------------------------------
__global__ void attn_apply_kernel(const float* __restrict__ attn,
                                  const float* __restrict__ vbuf,
                                  const float* __restrict__ edge_v,
                                  float* __restrict__ outbuf)
{
    const int bid = blockIdx.x;          // ((n*H + h)*S + i)
    const int i  = bid % cS;
    const int nh = bid / cS;
    const int h  = nh & 7;
    const int n  = nh >> 3;
    const int t  = threadIdx.x;          // 0..63

    __shared__ float p[cS];
    const size_t arow = ((size_t)nh * cS + i) * cS;
    p[t] = attn[arow + t];
    if (t < cS - cD) p[cD + t] = attn[arow + cD + t];

    const float* vp    = vbuf + (size_t)n * cE + h * cD + t;                     // j stride 4096
    const float* gbase = edge_v + ((size_t)(i * cS) * cN + n) * cE + h * cD;     // j stride 4096
    float acc = 0.0f;

#if defined(HAVE_ASYNC_LDS)
    __shared__ __align__(16) float evs[2][4 * cD];   // 2 stages x 4 rows x 64 floats

    // Stage copy layout: thread t -> row r = t>>4 (0..3), 16B chunk c = t&15.
    const int r = t >> 4;
    const int c = t & 15;
    const float* gsrc = gbase + (size_t)r * (cN * cE) + c * 4;

    typedef __attribute__((address_space(1))) v4i gv4i;   // global int4*
    typedef __attribute__((address_space(3))) v4i lv4i;   // LDS int4*

    // issue stage 0
    __builtin_amdgcn_global_load_async_to_lds_b128(
        (gv4i*)gsrc, (lv4i*)&evs[0][r * cD + c * 4], 0, 0);

    for (int jb = 0; jb < cS; jb += 4) {
        const int buf = (jb >> 2) & 1;
        if (jb + 4 < cS) {
            __builtin_amdgcn_global_load_async_to_lds_b128(
                (gv4i*)(gsrc + (size_t)(jb + 4) * (cN * cE)),
                (lv4i*)&evs[buf ^ 1][r * cD + c * 4], 0, 0);
            asm volatile("s_wait_asynccnt 0x1" ::: "memory");
        } else {
            asm volatile("s_wait_asynccnt 0x0" ::: "memory");
        }
        __syncthreads();                 // stage visible to both waves (also covers p[])
#pragma unroll
        for (int u = 0; u < 4; ++u) {
            const int j = jb + u;
            acc += p[j] * (vp[(size_t)j * (cN * cE)] + evs[buf][u * cD + t]);
        }
        __syncthreads();                 // all reads done before buffer refill
    }
#else
    __syncthreads();
    for (int j = 0; j < cS; ++j) {
        if (j + 4 < cS) __builtin_prefetch(gbase + (size_t)(j + 4) * (cN * cE) + t, 0, 1);
        acc += p[j] * (vp[(size_t)j * (cN * cE)] + gbase[(size_t)j * (cN * cE) + t]);
    }
#endif
    outbuf[((size_t)(i * cN + n)) * cE + h * cD + t] = acc;
}

// ---------------------------------------------------------------------------
// out[r,:] = LayerNorm(a[r,:] + b[r,:]) * gamma + beta, E=512, eps=1e-5.
// One block (128 threads, 4 elems each) per row.
// ---------------------------------------------------------------------------
__global__ void add_ln_kernel(const float* __restrict__ a, const float* __restrict__ b,
                              const float* __restrict__ gamma, const float* __restrict__ beta,
                              float* __restrict__ out)
{
    const int r = blockIdx.x;
    const int t = threadIdx.x;           // blockDim = 128
    const float* ar = a + (size_t)r * cE;
    const float* br = b + (size_t)r * cE;

    float x[4];
    float s = 0.0f;
#pragma unroll
    for (int u = 0; u < 4; ++u) {
        const int e = t + 128 * u;
        x[u] = ar[e] + br[e];
        s += x[u];
    }
    __shared__ float red[128];
    red[t] = s;
    __syncthreads();
    for (int off = 64; off > 0; off >>= 1) {
        if (t < off) red[t] += red[t + off];
        __syncthreads();
    }
    const float mu = red[0] * (1.0f / (float)cE);
    __syncthreads();
    float vs = 0.0f;
#pragma unroll
    for (int u = 0; u < 4; ++u) {
        const float dx = x[u] - mu;
        vs += dx * dx;
    }
    red[t] = vs;
    __syncthreads();
    for (int off = 64; off > 0; off >>= 1) {
        if (t < off) red[t] += red[t + off];
        __syncthreads();
    }
    const float rstd = rsqrtf(red[0] * (1.0f / (float)cE) + 1e-5f);
    __syncthreads();
#pragma unroll
    for (int u = 0; u < 4; ++u) {
        const int e = t + 128 * u;
        out[(size_t)r * cE + e] = (x[u] - mu) * rstd * gamma[e] + beta[e];
    }
}

// ---------------------------------------------------------------------------
extern "C" void kernel_launch(void* const* d_in, const int* in_sizes, int n_in,
                              void* d_out, int out_size, void* d_ws, size_t ws_size,
                              hipStream_t stream) {
    const float* src    = (const float*)d_in[0];
    const float* edge_k = (const float*)d_in[1];
    const float* edge_v = (const float*)d_in[2];
    const float* w_in   = (const float*)d_in[3];
    const float* b_in   = (const float*)d_in[4];
    const float* w_v    = (const float*)d_in[5];
    const float* b_v    = (const float*)d_in[6];
    const float* w_out  = (const float*)d_in[7];
    const float* b_out  = (const float*)d_in[8];
    const float* w1     = (const float*)d_in[9];
    const float* b1     = (const float*)d_in[10];
    const float* w2     = (const float*)d_in[11];
    const float* b2     = (const float*)d_in[12];
    const float* ln1_w  = (const float*)d_in[13];
    const float* ln1_b  = (const float*)d_in[14];
    const float* ln2_w  = (const float*)d_in[15];
    const float* ln2_b  = (const float*)d_in[16];
    const unsigned char* mask = (const unsigned char*)d_in[17];

    // Workspace layout (floats); total 4,915,200 floats = 19.66 MB
    float* ws   = (float*)d_ws;
    float* qk   = ws;                    // 768*1024
    float* vbuf = qk   + cM * 2 * cE;    // 768*512
    float* attn = vbuf + cM * cE;        // 64*96*96
    float* aout = attn + cN * cH * cS * cS; // 768*512
    float* proj = aout + cM * cE;        // 768*512
    float* x1   = proj + cM * cE;        // 768*512
    float* ffh  = x1   + cM * cE;        // 768*2048
    float* ffo  = ffh  + cM * cF;        // 768*512

    // 1) qk = src @ w_in^T + b_in              (768 x 1024, K=512)
    gemm_nt_wmma<<<(48 * 16) / 8, 256, 0, stream>>>(src, w_in, b_in, qk, cM, 2 * cE, cE, 0);
    // 2) v = src @ w_v^T + b_v                 (768 x 512, K=512)
    gemm_nt_wmma<<<(48 * 8) / 8, 256, 0, stream>>>(src, w_v, b_v, vbuf, cM, cE, cE, 0);
    // 3) dense logits q @ k^T                  (per (n,h): 96x96, K=64)
    qk_scores_wmma<<<(64 * 6) / 8, 256, 0, stream>>>(qk, attn);
    // 4) + q.ek edge logits, scale, mask, softmax
    edge_softmax_kernel<<<cN * cH * cS, 128, 0, stream>>>(qk, edge_k, mask, attn);
    // 5) attn @ (v + ev)
    attn_apply_kernel<<<cN * cH * cS, cD, 0, stream>>>(attn, vbuf, edge_v, aout);
    // 6) out projection
    gemm_nt_wmma<<<(48 * 8) / 8, 256, 0, stream>>>(aout, w_out, b_out, proj, cM, cE, cE, 0);
    // 7) x1 = LN(src + proj)
    add_ln_kernel<<<cM, 128, 0, stream>>>(src, proj, ln1_w, ln1_b, x1);
    // 8) ffh = relu(x1 @ w1^T + b1)            (768 x 2048, K=512)
    gemm_nt_wmma<<<(48 * 32) / 8, 256, 0, stream>>>(x1, w1, b1, ffh, cM, cF, cE, 1);
    // 9) ffo = ffh @ w2^T + b2                 (768 x 512, K=2048)
    gemm_nt_wmma<<<(48 * 8) / 8, 256, 0, stream>>>(ffh, w2, b2, ffo, cM, cE, cF, 0);
    // 10) out = LN(x1 + ffo)
    add_ln_kernel<<<cM, 128, 0, stream>>>(x1, ffo, ln2_w, ln2_b, (float*)d_out);
}